// WeightedEmbedding_48249662603573
// MI455X (gfx1250) — compile-verified
//
#include <hip/hip_runtime.h>

typedef __attribute__((ext_vector_type(2))) float v2f;
typedef __attribute__((ext_vector_type(4))) float v4f;
typedef __attribute__((ext_vector_type(8))) float v8f;

#define K_TOTAL 4096
#define KHALF   2048
#define N_TOTAL 32
#define CHUNK   128              // K-rows of weight per half staged per round
#define STR     (CHUNK + 4)      // padded k-stride (floats); even -> 8B align
#define ROUNDS  (KHALF / CHUNK)  // 16
#define GROUPS  (CHUNK / 32)     // 4 groups of 8 k-steps per round

// out[M,32] = x[M,4096] * w[4096,32],  M = 8192.
// Block = 128 threads = 4 waves = 2 (m-tile) pairs x 2 (K-half) waves;
// 256 blocks -> 2 blocks/WGP so one block's compute overlaps the other's
// weight staging. Each wave computes a PARTIAL 16x32 tile over half of K
// with v_wmma_f32_16x16x4_f32. A fragments double-buffered in registers;
// next round's x lines prefetched via global_prefetch; K-half partials
// combined through LDS.
__global__ __launch_bounds__(128)
void wgemm_f32_wmma_kernel(const float* __restrict__ x,
                           const float* __restrict__ w,
                           float* __restrict__ out) {
  // Weight chunk, transposed, both K-halves: ldsW[(c*2 + h)*STR + k]
  __shared__ float ldsW[N_TOTAL * 2 * STR];   // 8448 floats = 33 KB

  const int tid  = threadIdx.x;
  const int wave = tid >> 5;
  const int lane = tid & 31;
  const int half = lane >> 4;     // K sub-pair select within fp32 A/B layout
  const int lm   = lane & 15;

  const int pair = wave >> 1;     // m-tile within block (0..1)
  const int kh   = wave & 1;      // K half (0..1)
  const int tile_m = blockIdx.x * 2 + pair;

  // A: this lane covers row (tile_m*16 + lm), K range [kh*2048, +2048)
  const float* __restrict__ xk =
      x + (size_t)(tile_m * 16 + lm) * K_TOTAL + (size_t)kh * KHALF;
  // B: this lane covers weight columns lm (n-tile 0) and lm+16 (n-tile 1)
  const float* __restrict__ wn0 = &ldsW[(lm * 2 + kh) * STR];
  const float* __restrict__ wn1 = &ldsW[((lm + 16) * 2 + kh) * STR];

  v8f acc0 = {};   // n-tile 0
  v8f acc1 = {};   // n-tile 1

  v2f abuf[2][8];  // register double-buffer of A fragments

  // Issue 8 independent A loads for one group into distinct registers.
  auto loadA = [&](int buf, int base) {
    #pragma unroll
    for (int s = 0; s < 8; ++s)
      abuf[buf][s] = *(const v2f*)(xk + base + 4 * s + 2 * half);
  };
  // 16 WMMAs for one group; B fragments have a one-step register lookahead
  // so ds loads for step s+1 issue before step s's WMMAs.
  auto compute = [&](int buf, int base) {
    v2f b0 = *(const v2f*)(wn0 + base + 2 * half);
    v2f b1 = *(const v2f*)(wn1 + base + 2 * half);
    #pragma unroll
    for (int s = 0; s < 8; ++s) {
      v2f nb0, nb1;
      if (s < 7) {
        const int kn = base + 4 * (s + 1) + 2 * half;
        nb0 = *(const v2f*)(wn0 + kn);
        nb1 = *(const v2f*)(wn1 + kn);
      }
      acc0 = __builtin_amdgcn_wmma_f32_16x16x4_f32(
          false, abuf[buf][s], false, b0, (short)0, acc0, false, false);
      acc1 = __builtin_amdgcn_wmma_f32_16x16x4_f32(
          false, abuf[buf][s], false, b1, (short)0, acc1, false, false);
      if (s < 7) { b0 = nb0; b1 = nb1; }
    }
  };

  for (int r = 0; r < ROUNDS; ++r) {
    const int kc = r * CHUNK;

    // Prefetch A group 0 (global; overlaps weight staging + barrier).
    loadA(0, kc);

    // ---- stage transposed weight chunks (both K-halves) into LDS ----
    // 2*CHUNK*32 = 8192 floats = 2048 float4; 128 threads -> 16 b128 each.
    #pragma unroll
    for (int i = 0; i < (2 * CHUNK * N_TOTAL) / (128 * 4); ++i) {
      int idx = tid + i * 128;          // linear over (h, row, c4)
      int c4 = idx & 7;                 // float4 index within a 32-col row
      int t2 = idx >> 3;
      int rr = t2 & (CHUNK - 1);        // k within chunk
      int h  = t2 >> 7;                 // K half
      v4f q = *(const v4f*)(w + (size_t)(h * KHALF + kc + rr) * N_TOTAL + c4 * 4);
      #pragma unroll
      for (int j = 0; j < 4; ++j)
        ldsW[((c4 * 4 + j) * 2 + h) * STR + rr] = q[j];
    }
    __syncthreads();

    // ---- prefetch next round's x lines (global_prefetch_b8: no LOADcnt,
    // no VGPR dest). Each prefetch covers this lane's row, one 128B line.
    if (r + 1 < ROUNDS) {
      #pragma unroll
      for (int p = 0; p < CHUNK / 32; ++p)
        __builtin_prefetch(xk + kc + CHUNK + p * 32 + 2 * half, 0, 3);
    }

    // ---- 4 groups, software-pipelined: load group g+1 while computing g
    #pragma unroll
    for (int g = 0; g < GROUPS; ++g) {
      if (g + 1 < GROUPS) loadA((g + 1) & 1, kc + (g + 1) * 32);
      compute(g & 1, kc + g * 32);
    }
    __syncthreads();
  }

  // ---- combine K-half partials through LDS, kh==0 waves store ----
  float* __restrict__ red = ldsW;     // reuse staging LDS (safe after barrier)
  const int rbase = (pair * 32 + lane) * 16;
  if (kh == 1) {
    #pragma unroll
    for (int j = 0; j < 8; ++j) {
      red[rbase + j]     = acc0[j];
      red[rbase + 8 + j] = acc1[j];
    }
  }
  __syncthreads();
  if (kh == 0) {
    // C/D layout: VGPR j -> (M = j + 8*half, N = lane&15)
    float* __restrict__ orow = out + (size_t)(tile_m * 16) * N_TOTAL + lm;
    #pragma unroll
    for (int j = 0; j < 8; ++j) {
      float s0 = acc0[j] + red[rbase + j];
      float s1 = acc1[j] + red[rbase + 8 + j];
      orow[(size_t)(j + 8 * half) * N_TOTAL]      = s0;   // n-tile 0
      orow[(size_t)(j + 8 * half) * N_TOTAL + 16] = s1;   // n-tile 1
    }
  }
}

extern "C" void kernel_launch(void* const* d_in, const int* in_sizes, int n_in,
                              void* d_out, int out_size, void* d_ws, size_t ws_size,
                              hipStream_t stream) {
  (void)in_sizes; (void)n_in; (void)d_ws; (void)ws_size; (void)out_size;
  const float* x = (const float*)d_in[0];   // (4, 2048, 4096) f32
  const float* w = (const float*)d_in[1];   // (4096, 32) f32
  float* out = (float*)d_out;               // (4, 2048, 32) f32

  const int M = 4 * 2048;                   // 8192 rows
  const int rows_per_block = 32;            // 2 m-tiles per block
  dim3 grid(M / rows_per_block);            // 256 blocks, 1024 waves
  dim3 block(128);
  hipLaunchKernelGGL(wgemm_f32_wmma_kernel, grid, block, 0, stream, x, w, out);
}